// ContrastiveLoss_14096082666269
// MI455X (gfx1250) — compile-verified
//
#include <hip/hip_runtime.h>
#include <hip/hip_bf16.h>
#include <math.h>

#define N_CLASSES 19
#define KSEG 20              // N_CLASSES + 1 (ignore bin)
#define HW (512*512)
#define BATCH 16
#define TAU_INV 10.0f

#define SUMS_ELEMS (BATCH*KSEG*N_CLASSES)
#define CNTS_ELEMS (BATCH*KSEG)

typedef __attribute__((ext_vector_type(2))) float v2f;
typedef __attribute__((ext_vector_type(8))) float v8f;

// ---------------------------------------------------------------- zero scratch
__global__ void zero_ws(float* __restrict__ ws, float* __restrict__ out, int out_n) {
    int i = blockIdx.x * blockDim.x + threadIdx.x;
    if (i < SUMS_ELEMS + CNTS_ELEMS) ws[i] = 0.0f;
    if (i < out_n) out[i] = 0.0f;
}

// ------------------------------------------------- fused normalize + segment sum
#define TPB 256
#define PPT 8                 // pixels per thread (2 float4 iterations)
#define NHALF 16              // per-half-wave private bins (16-lane groups)

__global__ __launch_bounds__(TPB) void accum_kernel(
    const float* __restrict__ q,      // [B][C][H][W]
    const int*   __restrict__ labels, // [B][H][W]
    float*       __restrict__ ws)     // sums then counts
{
    __shared__ float acc[NHALF][KSEG][21];   // slot 19 = count, stride 21 => 20 distinct banks
    const int tid  = threadIdx.x;
    const int grp  = tid >> 4;               // half-wave id 0..15
    const int b    = blockIdx.y;

    for (int i = tid; i < NHALF*KSEG*21; i += TPB) ((float*)acc)[i] = 0.0f;
    __syncthreads();

    const size_t qbase = (size_t)b * N_CLASSES * HW;
    const int*   lb    = labels + (size_t)b * HW;
    const int    p4base = blockIdx.x * (TPB * PPT / 4);

    for (int it = 0; it < PPT/4; ++it) {
        const int p4 = p4base + it * TPB + tid;   // float4 index, coalesced
        float v[4][N_CLASSES];
        #pragma unroll
        for (int c = 0; c < N_CLASSES; ++c) {
            float4 t = reinterpret_cast<const float4*>(q + qbase + (size_t)c * HW)[p4];
            v[0][c] = t.x; v[1][c] = t.y; v[2][c] = t.z; v[3][c] = t.w;
        }
        const int4 l4 = reinterpret_cast<const int4*>(lb)[p4];
        const int lab[4] = {l4.x, l4.y, l4.z, l4.w};

        #pragma unroll
        for (int j = 0; j < 4; ++j) {
            float ss = 0.0f;
            #pragma unroll
            for (int c = 0; c < N_CLASSES; ++c) ss += v[j][c] * v[j][c];
            const float scale = 1.0f / fmaxf(sqrtf(ss), 1e-12f);
            int l = lab[j]; l = (l == 255) ? N_CLASSES : l;
            float* bin = &acc[grp][l][0];
            #pragma unroll
            for (int c = 0; c < N_CLASSES; ++c) atomicAdd(&bin[c], v[j][c] * scale);
            atomicAdd(&bin[N_CLASSES], 1.0f);
        }
    }
    __syncthreads();

    // block flush: 20 labels x (19 sums + 1 count), fixed-order over the 16 copies
    for (int i = tid; i < KSEG * 20; i += TPB) {
        const int l = i / 20, e = i % 20;
        float s = 0.0f;
        #pragma unroll
        for (int g = 0; g < NHALF; ++g) s += acc[g][l][e];
        if (e < N_CLASSES)
            atomicAdd(&ws[((size_t)b * KSEG + l) * N_CLASSES + e], s);
        else
            atomicAdd(&ws[SUMS_ELEMS + b * KSEG + l], s);
    }
}

// --------------------------------------------- finalize: WMMA logits + CE + mean
__global__ __launch_bounds__(32) void finalize_kernel(
    const float* __restrict__ emb_k,  // [19][19]
    const float* __restrict__ ws,
    const int*   __restrict__ epoch,
    float*       __restrict__ out)
{
    const int b    = blockIdx.x;
    const int lane = threadIdx.x;
    const float* sums   = ws;
    const float* counts = ws + SUMS_ELEMS;

    __shared__ float meansP[32][20];   // A: rows m(=class) padded to 32, K(=channel) padded to 20
    __shared__ float bP[20][32];       // B: [K=channel][N=key] normalized keys transposed
    __shared__ float logits[32][33];
    __shared__ float rnorm[32];
    __shared__ float red[2][32];

    // inverse row-norms of emb_k
    {
        float rn = 0.0f;
        if (lane < N_CLASSES) {
            float ss = 0.0f;
            for (int c = 0; c < N_CLASSES; ++c) { float t = emb_k[lane * N_CLASSES + c]; ss += t * t; }
            rn = 1.0f / fmaxf(sqrtf(ss), 1e-12f);
        }
        rnorm[lane] = rn;
    }
    __syncthreads();

    for (int idx = lane; idx < 32 * 20; idx += 32) {       // A tile (zero-padded)
        const int m = idx / 20, c = idx % 20;
        float val = 0.0f;
        if (m < KSEG && c < N_CLASSES) {
            const float cnt = counts[b * KSEG + m];
            val = sums[((size_t)b * KSEG + m) * N_CLASSES + c] / fmaxf(cnt, 1.0f);
        }
        meansP[m][c] = val;
    }
    for (int idx = lane; idx < 20 * 32; idx += 32) {       // B tile (zero-padded)
        const int c = idx / 32, n = idx % 32;
        float val = 0.0f;
        if (c < N_CLASSES && n < N_CLASSES)
            val = emb_k[n * N_CLASSES + c] * rnorm[n];
        bP[c][n] = val;
    }
    __syncthreads();

    // logits = means @ ekn^T via chained V_WMMA_F32_16X16X4_F32 (exact fp32)
    // A 16x4 layout: lanes 0-15 -> M=lane, VGPR0=K0/VGPR1=K1; lanes 16-31 -> K2/K3.
    // B 4x16: lanes 0-15 -> N=lane, VGPR0=K0/VGPR1=K1; lanes 16-31 -> K2/K3.
    // D 16x16: VGPR r -> M=r (+8 for lanes 16-31), N=lane&15.
    const int half = lane >> 4;
    const int ln   = lane & 15;
    #pragma unroll
    for (int mt = 0; mt < 2; ++mt) {
        #pragma unroll
        for (int nt = 0; nt < 2; ++nt) {
            v8f accv = {};
            #pragma unroll
            for (int kk = 0; kk < 5; ++kk) {
                const int k0 = kk * 4 + half * 2;
                v2f av; av.x = meansP[mt * 16 + ln][k0]; av.y = meansP[mt * 16 + ln][k0 + 1];
                v2f bv; bv.x = bP[k0][nt * 16 + ln];     bv.y = bP[k0 + 1][nt * 16 + ln];
                accv = __builtin_amdgcn_wmma_f32_16x16x4_f32(
                    false, av, false, bv, (short)0, accv, false, false);
            }
            #pragma unroll
            for (int r = 0; r < 8; ++r)
                logits[mt * 16 + half * 8 + r][nt * 16 + ln] = accv[r];
        }
    }
    __syncthreads();

    // per-class log-softmax CE (lane = class row)
    float cv = 0.0f, vv = 0.0f;
    if (lane < N_CLASSES) {
        float mx = -3.4e38f;
        for (int n = 0; n < N_CLASSES; ++n) mx = fmaxf(mx, logits[lane][n] * TAU_INV);
        float se = 0.0f;
        for (int n = 0; n < N_CLASSES; ++n) se += expf(logits[lane][n] * TAU_INV - mx);
        const float lse = mx + logf(se);
        const float ce  = -(logits[lane][lane] * TAU_INV - lse);
        vv = (counts[b * KSEG + lane] > 0.0f) ? 1.0f : 0.0f;
        cv = ce * vv;
    }
    red[0][lane] = cv; red[1][lane] = vv;
    __syncthreads();

    if (lane == 0) {
        float s = 0.0f, nv = 0.0f;
        for (int i = 0; i < N_CLASSES; ++i) { s += red[0][i]; nv += red[1][i]; }
        const float per = s / fmaxf(nv, 1.0f);
        float contrib = (nv > 0.0f) ? per : 0.0f;
        if (*epoch == 0) contrib = 0.0f;
        atomicAdd(out, contrib * (1.0f / BATCH));
    }
}

// ---------------------------------------------------------------------- launch
extern "C" void kernel_launch(void* const* d_in, const int* in_sizes, int n_in,
                              void* d_out, int out_size, void* d_ws, size_t ws_size,
                              hipStream_t stream) {
    const float* emb_k  = (const float*)d_in[0];   // (19, 19) f32
    const float* emb_q  = (const float*)d_in[1];   // (16, 19, 512, 512) f32
    const int*   labels = (const int*)d_in[2];     // (16, 512, 512) i32
    const int*   epoch  = (const int*)d_in[3];     // scalar i32
    float* out = (float*)d_out;
    float* ws  = (float*)d_ws;

    const int zn = SUMS_ELEMS + CNTS_ELEMS;        // 6400 floats of scratch
    zero_ws<<<(zn + 255) / 256, 256, 0, stream>>>(ws, out, out_size);

    dim3 grid(HW / (TPB * PPT), BATCH);            // (128, 16)
    accum_kernel<<<grid, TPB, 0, stream>>>(emb_q, labels, ws);

    finalize_kernel<<<BATCH, 32, 0, stream>>>(emb_k, ws, epoch, out);
}